// CalculateInstanceSize_36807869727201
// MI455X (gfx1250) — compile-verified
//
#include <hip/hip_runtime.h>
#include <math.h>

#define Hh 512
#define Ww 512
#define Bb 8
#define Nn 32

typedef __attribute__((ext_vector_type(2))) float v2f;
typedef __attribute__((ext_vector_type(8))) float v8f;

__device__ __forceinline__ float wave_sum_f(float v) {
    #pragma unroll
    for (int off = 16; off > 0; off >>= 1) v += __shfl_xor(v, off, 32);
    return v;
}

// ---------------------------------------------------------------------------
// Kernel A: per-image road fit -> unit[b][h]  (one block per image)
// ---------------------------------------------------------------------------
__global__ __launch_bounds__(256) void unit_kernel(const float* __restrict__ seg,
                                                   float* __restrict__ unitOut) {
    __shared__ int s_xmin[Hh];
    __shared__ int s_xmax[Hh];
    __shared__ unsigned s_words[16];
    __shared__ float s_part[8][7];
    __shared__ float s_theta[4];

    const int b    = blockIdx.x;
    const int t    = threadIdx.x;
    const int wv   = t >> 5;
    const int lane = t & 31;

    const float* img = seg + (size_t)b * Hh * Ww * 2;  // channel stride 2, use +1

    // per-row min/max of positive columns; wave wv handles rows wv + 8*i
    for (int i = 0; i < Hh / 8; ++i) {
        const int h = wv + 8 * i;
        const float* row = img + (size_t)h * Ww * 2;
        int lmin = Ww, lmax = -1;
        #pragma unroll
        for (int j = 0; j < Ww / 32; ++j) {
            const int w = lane + 32 * j;
            const float val = row[w * 2 + 1];
            if (val > 0.0f) { lmin = min(lmin, w); lmax = max(lmax, w); }
        }
        #pragma unroll
        for (int off = 16; off > 0; off >>= 1) {
            lmin = min(lmin, __shfl_xor(lmin, off, 32));
            lmax = max(lmax, __shfl_xor(lmax, off, 32));
        }
        if (lane == 0) { s_xmin[h] = lmin; s_xmax[h] = lmax; }
    }
    __syncthreads();

    // validity bitmap (512 bits in 16 words) via wave ballots
    const int hA = t, hB = t + 256;
    const bool vA = (s_xmin[hA] != Ww) && (s_xmin[hA] != s_xmax[hA]);
    const bool vB = (s_xmin[hB] != Ww) && (s_xmin[hB] != s_xmax[hB]);
    const unsigned mA = __builtin_amdgcn_ballot_w32(vA);
    const unsigned mB = __builtin_amdgcn_ballot_w32(vB);
    if (lane == 0) { s_words[wv] = mA; s_words[wv + 8] = mB; }
    __syncthreads();

    unsigned words[16];
    int n_valid = 0;
    #pragma unroll
    for (int k = 0; k < 16; ++k) { words[k] = s_words[k]; n_valid += __popc(words[k]); }
    int drop = (int)((float)n_valid * 0.15f);   // truncation, like astype(int32)
    if (drop < 1) drop = 1;
    const int hiLim = n_valid - drop;

    // weighted partial sums over this thread's two rows
    float Sw = 0, Sy = 0, Syy = 0, Sxl = 0, Syxl = 0, Sxr = 0, Syxr = 0;
    #pragma unroll
    for (int r = 0; r < 2; ++r) {
        const int h = (r == 0) ? hA : hB;
        const bool valid = (r == 0) ? vA : vB;
        if (valid) {
            const int k = h >> 5, bit = h & 31;
            int rank = 0;
            for (int j = 0; j < k; ++j) rank += __popc(words[j]);
            rank += __popc(words[k] & ((bit == 0) ? 0u : ((1u << bit) - 1u)));
            if (rank >= drop && rank < hiLim) {
                const float y  = (float)h;
                const float xl = (float)s_xmin[h];
                const float xr = (float)s_xmax[h];
                Sw += 1.0f; Sy += y; Syy += y * y;
                Sxl += xl; Syxl += y * xl;
                Sxr += xr; Syxr += y * xr;
            }
        }
    }
    Sw = wave_sum_f(Sw);   Sy = wave_sum_f(Sy);   Syy = wave_sum_f(Syy);
    Sxl = wave_sum_f(Sxl); Syxl = wave_sum_f(Syxl);
    Sxr = wave_sum_f(Sxr); Syxr = wave_sum_f(Syxr);
    if (lane == 0) {
        s_part[wv][0] = Sw;  s_part[wv][1] = Sy;  s_part[wv][2] = Syy;
        s_part[wv][3] = Sxl; s_part[wv][4] = Syxl;
        s_part[wv][5] = Sxr; s_part[wv][6] = Syxr;
    }
    __syncthreads();

    if (t == 0) {
        float a[7] = {0, 0, 0, 0, 0, 0, 0};
        for (int wvi = 0; wvi < 8; ++wvi)          // fixed order -> deterministic
            for (int q = 0; q < 7; ++q) a[q] += s_part[wvi][q];
        const float det  = a[2] * a[0] - a[1] * a[1];   // Syy*Sw - Sy*Sy
        const float safe = (det > 0.0f) ? det : 1.0f;
        const bool  ok   = det > 0.0f;
        const float sl_l = (a[0] * a[4] - a[1] * a[3]) / safe;
        const float ic_l = (-a[1] * a[4] + a[2] * a[3]) / safe;
        const float sl_r = (a[0] * a[6] - a[1] * a[5]) / safe;
        const float ic_r = (-a[1] * a[6] + a[2] * a[5]) / safe;
        s_theta[0] = ok ? sl_l : 0.0f; s_theta[1] = ok ? ic_l : 0.0f;
        s_theta[2] = ok ? sl_r : 0.0f; s_theta[3] = ok ? ic_r : 0.0f;
    }
    __syncthreads();

    const float sl_l = s_theta[0], ic_l = s_theta[1];
    const float sl_r = s_theta[2], ic_r = s_theta[3];
    #pragma unroll
    for (int r = 0; r < 2; ++r) {
        const int h = (r == 0) ? hA : hB;
        const float y = (float)h;
        float width = (y * sl_r + ic_r) - (y * sl_l + ic_l);
        width = fmaxf(width, 1.0f);
        unitOut[b * Hh + h] = 3.25f / width;
    }
}

// ---------------------------------------------------------------------------
// Kernel B: fused single-pass reduction over pad[b][n][H][W].
// One block per (b,n). float4 loads per lane feed 4 interleaved "virtual
// chunks" of V_WMMA_F32_16X16X4_F32 (column permutation is legal: all three
// outputs are column-order invariant). Each element is touched exactly once:
// the matrix pipe accumulates colsum (for horizontal max), the VALU does the
// unit^2 fmaf (instance) and a max4+compare occupancy bit (vertical).
// ---------------------------------------------------------------------------
__global__ __launch_bounds__(256) void reduce_kernel(const float* __restrict__ pad,
                                                     const float* __restrict__ unitIn,
                                                     float* __restrict__ out) {
    __shared__ float s_unit[Hh];
    __shared__ float s_colsum[8][Ww];
    __shared__ float s_red[2][8];
    __shared__ float s_max[8];

    const int bid  = blockIdx.x;
    const int b    = bid >> 5;   // / N (N=32)
    const int n    = bid & 31;
    const int t    = threadIdx.x;
    const int wv   = t >> 5;
    const int lane = t & 31;
    const int half = lane >> 4;  // 0: lanes 0-15, 1: lanes 16-31
    const int ln   = lane & 15;  // WMMA n-index within a chunk

    s_unit[t]       = unitIn[b * Hh + t];
    s_unit[t + 256] = unitIn[b * Hh + t + 256];
    __syncthreads();

    const float* img = pad + (size_t)(b * Nn + n) * Hh * Ww;

    float inst = 0.0f;
    unsigned occ0 = 0u, occ1 = 0u;   // per row-group occupancy bits (16 groups/wave)

    // wave wv owns row-groups g = wv + 8*i (i=0..15), each 4 rows tall.
    // Columns: 8 sets of 64; lane ln covers cols base + 4*ln + q, q=0..3.
    for (int set = 0; set < 8; ++set) {
        const int base = set * 64;
        v8f c0 = {0.f, 0.f, 0.f, 0.f, 0.f, 0.f, 0.f, 0.f};
        v8f c1 = {0.f, 0.f, 0.f, 0.f, 0.f, 0.f, 0.f, 0.f};
        v8f c2 = {0.f, 0.f, 0.f, 0.f, 0.f, 0.f, 0.f, 0.f};
        v8f c3 = {0.f, 0.f, 0.f, 0.f, 0.f, 0.f, 0.f, 0.f};
        #pragma unroll 4
        for (int i = 0; i < 16; ++i) {
            const int h0 = 4 * (wv + 8 * i);
            const int r0 = h0 + 2 * half;    // B layout: v0 rows K=0/K=2 per half
            const int r1 = r0 + 1;           //           v1 rows K=1/K=3 per half
            const float4 x0 = *(const float4*)(img + (size_t)r0 * Ww + base + 4 * ln);
            const float4 x1 = *(const float4*)(img + (size_t)r1 * Ww + base + 4 * ln);
            const float a0 = s_unit[r0];
            const float a1 = s_unit[r1];
            inst = fmaf(a0 * a0, (x0.x + x0.y) + (x0.z + x0.w), inst);
            inst = fmaf(a1 * a1, (x1.x + x1.y) + (x1.z + x1.w), inst);
            const float m0 = fmaxf(fmaxf(x0.x, x0.y), fmaxf(x0.z, x0.w));
            const float m1 = fmaxf(fmaxf(x1.x, x1.y), fmaxf(x1.z, x1.w));
            occ0 |= (m0 > 0.5f) ? (1u << i) : 0u;   // any>0.5 == max4>0.5
            occ1 |= (m1 > 0.5f) ? (1u << i) : 0u;
            const v2f A  = {a0, a1};                 // A[m][k]=unit[h0+k] for all m
            const v2f b0v = {x0.x, x1.x};
            const v2f b1v = {x0.y, x1.y};
            const v2f b2v = {x0.z, x1.z};
            const v2f b3v = {x0.w, x1.w};
            c0 = __builtin_amdgcn_wmma_f32_16x16x4_f32(false, A, false, b0v, (short)0, c0, false, false);
            c1 = __builtin_amdgcn_wmma_f32_16x16x4_f32(false, A, false, b1v, (short)0, c1, false, false);
            c2 = __builtin_amdgcn_wmma_f32_16x16x4_f32(false, A, false, b2v, (short)0, c2, false, false);
            c3 = __builtin_amdgcn_wmma_f32_16x16x4_f32(false, A, false, b3v, (short)0, c3, false, false);
        }
        // every C row == partial colsum; row 0 lives in VGPR0, lanes 0-15.
        // q-th accumulator belongs to column base + 4*ln + q -> one b128 store.
        if (half == 0) {
            const float4 cs = make_float4(c0[0], c1[0], c2[0], c3[0]);
            *(float4*)(&s_colsum[wv][base + 4 * ln]) = cs;
        }
    }

    // vertical: resolve occupancy per row via half-wave ballots
    float vert = 0.0f;
    #pragma unroll
    for (int i = 0; i < 16; ++i) {
        const int h0 = 4 * (wv + 8 * i);
        const unsigned b0 = __builtin_amdgcn_ballot_w32((occ0 >> i) & 1u);
        const unsigned b1 = __builtin_amdgcn_ballot_w32((occ1 >> i) & 1u);
        if (lane == 0) {
            if (b0 & 0xFFFFu) vert += s_unit[h0];
            if (b0 >> 16)     vert += s_unit[h0 + 2];
            if (b1 & 0xFFFFu) vert += s_unit[h0 + 1];
            if (b1 >> 16)     vert += s_unit[h0 + 3];
        }
    }

    inst = wave_sum_f(inst);
    if (lane == 0) { s_red[0][wv] = inst; s_red[1][wv] = vert; }
    __syncthreads();

    // combine 8 wave colsum partials (fixed order) and take running max
    float m = -INFINITY;
    #pragma unroll
    for (int r = 0; r < 2; ++r) {
        const int w = t + 256 * r;
        float s = 0.0f;
        #pragma unroll
        for (int wvi = 0; wvi < 8; ++wvi) s += s_colsum[wvi][w];
        m = fmaxf(m, s);
    }
    #pragma unroll
    for (int off = 16; off > 0; off >>= 1) m = fmaxf(m, __shfl_xor(m, off, 32));
    if (lane == 0) s_max[wv] = m;
    __syncthreads();

    if (t == 0) {
        float instT = 0.0f, vertT = 0.0f, mx = s_max[0];
        for (int wvi = 0; wvi < 8; ++wvi) {       // fixed order -> deterministic
            instT += s_red[0][wvi];
            vertT += s_red[1][wvi];
            if (wvi) mx = fmaxf(mx, s_max[wvi]);
        }
        float* o = out + (size_t)bid * 3;
        o[0] = instT;   // instance_size
        o[1] = mx;      // horizontal_size
        o[2] = vertT;   // vertical_size
    }
}

extern "C" void kernel_launch(void* const* d_in, const int* in_sizes, int n_in,
                              void* d_out, int out_size, void* d_ws, size_t ws_size,
                              hipStream_t stream) {
    const float* seg = (const float*)d_in[0];   // [8,512,512,2] f32
    const float* pad = (const float*)d_in[1];   // [8,32,512,512] f32
    float* outp = (float*)d_out;                // [8,32,3] f32
    float* unit = (float*)d_ws;                 // 8*512 f32 scratch

    unit_kernel<<<Bb, 256, 0, stream>>>(seg, unit);
    reduce_kernel<<<Bb * Nn, 256, 0, stream>>>(pad, unit, outp);
}